// TetraDMPNN_82746839924783
// MI455X (gfx1250) — compile-verified
//
#include <hip/hip_runtime.h>

#ifdef __HIP_DEVICE_COMPILE__
#if !__has_builtin(__builtin_amdgcn_wmma_f32_16x16x4_f32)
#error "__builtin_amdgcn_wmma_f32_16x16x4_f32 not available on this toolchain"
#endif
#endif

typedef __attribute__((ext_vector_type(2))) float v2f;
typedef __attribute__((ext_vector_type(8))) float v8f;

#define N_NODES  50000
#define E_EDGES  200000
#define HDIM     128
#define FN_DIM   48
#define FE_DIM   12
#define T_TET    5000
#define G_GRAPH  500
#define NDEPTH   4
#define BN_EPS_F 1e-5f

// LDS geometry
#define LDA   132   // A tile stride (floats), K=128 kernels
#define LDA_I 68    // A tile stride (floats), K=64 (edge-init)
#define LDWP  144   // W pair-row stride (float2 units): 288*p mod 64 banks = 32*p
                    // -> kg=0 and kg=1 lanes use disjoint bank halves

// ---------------------------------------------------------------------------
// Block = 64 threads = 2 waves. Each wave owns 32 rows (two 16-row m-tiles)
// x 128 cols (8 n-tiles) of the 64-row block -> every B fragment feeds 2 WMMA.
// W is staged k-pair-interleaved: Wp[p][n] = {W[kb+2p][n], W[kb+2p+1][n]}
// so a B fragment is ONE aligned ds_load_b64 (no v_mov pair assembly).
// f32 WMMA 16x16x4 layout:
//   A frag: lane l -> row m=l&15, vgpr0=k0+2*(l>>4), vgpr1=+1
//   B frag: lane l -> col n=l&15, vgpr0=k0+2*(l>>4), vgpr1=+1
//   C/D:    lane l -> col n=l&15 (+16*nt), vgpr r -> row r+8*(l>>4)
// ---------------------------------------------------------------------------
__device__ __forceinline__ void wmma_kchunk2(const float* __restrict__ Alds,
                                             const v2f* __restrict__ Wp,
                                             int w, int lane, int kbase, int lda,
                                             v8f acc0[8], v8f acc1[8]) {
  const int m  = lane & 15;
  const int kg = lane >> 4;
#pragma unroll
  for (int kk = 0; kk < 16; kk += 4) {
    const int c = kbase + kk + 2 * kg;
    v2f a0 = *(const v2f*)(Alds + (w * 32 + m) * lda + c);
    v2f a1 = *(const v2f*)(Alds + (w * 32 + 16 + m) * lda + c);
    const int prow = (kk >> 1) + kg;  // pair-row within 16-k chunk (0..7)
#pragma unroll
    for (int nt = 0; nt < 8; ++nt) {
      v2f bf = Wp[prow * LDWP + nt * 16 + m];
      acc0[nt] = __builtin_amdgcn_wmma_f32_16x16x4_f32(
          false, a0, false, bf, (short)0, acc0[nt], false, false);
      acc1[nt] = __builtin_amdgcn_wmma_f32_16x16x4_f32(
          false, a1, false, bf, (short)0, acc1[nt], false, false);
    }
  }
}

__device__ __forceinline__ void acc_zero2(v8f acc0[8], v8f acc1[8]) {
#pragma unroll
  for (int nt = 0; nt < 8; ++nt)
#pragma unroll
    for (int i = 0; i < 8; ++i) {
      acc0[nt][i] = 0.0f;
      acc1[nt][i] = 0.0f;
    }
}

// Stage 16 k-rows (8 pair-rows) of KxH row-major W into pair-interleaved LDS.
__device__ __forceinline__ void stage_wp(const float* __restrict__ Wg, int kbase,
                                         int krows, v2f* __restrict__ Wp,
                                         int tid) {
#pragma unroll 4
  for (int i = tid; i < 8 * HDIM; i += 64) {
    int p = i >> 7, n = i & 127;
    int k0 = kbase + 2 * p;
    v2f v;
    v.x = (k0 < krows)     ? Wg[(size_t)k0 * HDIM + n]       : 0.0f;
    v.y = (k0 + 1 < krows) ? Wg[(size_t)(k0 + 1) * HDIM + n] : 0.0f;
    Wp[p * LDWP + n] = v;
  }
}

// K main-loop with double-buffered W chunks: stage chunk c+1 while chunk c
// feeds the matrix pipe.
__device__ __forceinline__ void gemm_main(const float* __restrict__ Alds,
                                          const float* __restrict__ Wg,
                                          int kdim, int krows, int lda,
                                          v2f* __restrict__ wp0,
                                          v2f* __restrict__ wp1, int tid, int w,
                                          int lane, v8f acc0[8], v8f acc1[8]) {
  stage_wp(Wg, 0, krows, wp0, tid);
  __syncthreads();
  const int nch = kdim >> 4;
  for (int c = 0; c < nch; ++c) {
    v2f* cur = (c & 1) ? wp1 : wp0;
    v2f* nxt = (c & 1) ? wp0 : wp1;
    if (c + 1 < nch) stage_wp(Wg, (c + 1) * 16, krows, nxt, tid);
    wmma_kchunk2(Alds, cur, w, lane, c * 16, lda, acc0, acc1);
    __syncthreads();
  }
}

// ---------------------------------------------------------------------------
// 1) ea = relu(concat(x[row], edge_attr) @ ei_W + ei_b)    (E,60)x(60,128)
// ---------------------------------------------------------------------------
__global__ __launch_bounds__(64) void edge_init_kernel(
    const float* __restrict__ x, const float* __restrict__ eattr,
    const int* __restrict__ rowA, const float* __restrict__ W,
    const float* __restrict__ b, float* __restrict__ out) {
  __shared__ __align__(16) float Alds[64 * LDA_I];
  __shared__ __align__(16) v2f WpBuf[2][8 * LDWP];
  const int tid = threadIdx.x, lane = tid & 31, w = tid >> 5;
  const int e0 = blockIdx.x * 64;

  for (int i = tid; i < 64 * 64; i += 64) {
    int r = i >> 6, c = i & 63;
    int e = e0 + r;
    float v = 0.0f;
    if (c < FN_DIM)               v = x[(size_t)rowA[e] * FN_DIM + c];
    else if (c < FN_DIM + FE_DIM) v = eattr[(size_t)e * FE_DIM + (c - FN_DIM)];
    Alds[r * LDA_I + c] = v;
  }
  __syncthreads();

  v8f acc0[8], acc1[8];
  acc_zero2(acc0, acc1);
  gemm_main(Alds, W, 64, FN_DIM + FE_DIM, LDA_I, WpBuf[0], WpBuf[1], tid, w,
            lane, acc0, acc1);

  const int mb = (lane >> 4) * 8;
  const int nc = lane & 15;
  float bias[8];
#pragma unroll
  for (int nt = 0; nt < 8; ++nt) bias[nt] = b[nt * 16 + nc];
#pragma unroll
  for (int mt = 0; mt < 2; ++mt) {
    const v8f* acc = mt ? acc1 : acc0;
#pragma unroll
    for (int r = 0; r < 8; ++r) {
      int e = e0 + w * 32 + mt * 16 + mb + r;
      float* orow = out + (size_t)e * HDIM;
#pragma unroll
      for (int nt = 0; nt < 8; ++nt)
        orow[nt * 16 + nc] = fmaxf(acc[nt][r] + bias[nt], 0.0f);
    }
  }
}

// ---------------------------------------------------------------------------
// 2) m = relu(h @ lin_W + lin_b); a[col[e]] += m[e]   (fused segment-sum)
// ---------------------------------------------------------------------------
__global__ __launch_bounds__(64) void msg_scatter_kernel(
    const float* __restrict__ h, const float* __restrict__ W,
    const float* __restrict__ b, const int* __restrict__ colA,
    float* __restrict__ a_out) {
  __shared__ __align__(16) float Alds[64 * LDA];
  __shared__ __align__(16) v2f WpBuf[2][8 * LDWP];
  const int tid = threadIdx.x, lane = tid & 31, w = tid >> 5;
  const int e0 = blockIdx.x * 64;

  for (int i = tid; i < 64 * (HDIM / 4); i += 64) {
    int r = i >> 5, c4 = (i & 31) << 2;
    *(float4*)(Alds + r * LDA + c4) =
        *(const float4*)(h + (size_t)(e0 + r) * HDIM + c4);
  }
  __syncthreads();

  v8f acc0[8], acc1[8];
  acc_zero2(acc0, acc1);
  gemm_main(Alds, W, HDIM, HDIM, LDA, WpBuf[0], WpBuf[1], tid, w, lane, acc0,
            acc1);

  const int mb = (lane >> 4) * 8;
  const int nc = lane & 15;
  float bias[8];
#pragma unroll
  for (int nt = 0; nt < 8; ++nt) bias[nt] = b[nt * 16 + nc];
#pragma unroll
  for (int mt = 0; mt < 2; ++mt) {
    const v8f* acc = mt ? acc1 : acc0;
#pragma unroll
    for (int r = 0; r < 8; ++r) {
      int e = e0 + w * 32 + mt * 16 + mb + r;
      float* abase = a_out + (size_t)colA[e] * HDIM;
#pragma unroll
      for (int nt = 0; nt < 8; ++nt) {
        float v = fmaxf(acc[nt][r] + bias[nt], 0.0f);
        atomicAdd(abase + nt * 16 + nc, v);
      }
    }
  }
}

// ---------------------------------------------------------------------------
// 3) tetra pairwise-difference product + signed sixth root (elementwise)
// ---------------------------------------------------------------------------
__global__ void tetra_t_kernel(const float* __restrict__ h,
                               const int* __restrict__ te,
                               float* __restrict__ tout) {
  int idx = blockIdx.x * blockDim.x + threadIdx.x;
  if (idx >= T_TET * HDIM) return;
  int t = idx >> 7, c = idx & 127;
  const int4 e4 = *(const int4*)(te + t * 4);
  float x0 = h[(size_t)e4.x * HDIM + c];
  float x1 = h[(size_t)e4.y * HDIM + c];
  float x2 = h[(size_t)e4.z * HDIM + c];
  float x3 = h[(size_t)e4.w * HDIM + c];
  float p = (x0 - x1) * (x0 - x2) * (x0 - x3) *
            (x1 - x2) * (x1 - x3) * (x2 - x3);
  float s = (p > 0.0f) ? 1.0f : ((p < 0.0f) ? -1.0f : 0.0f);
  tout[idx] = s * powf(fabsf(p) + 1e-6f, 1.0f / 6.0f);
}

// ---------------------------------------------------------------------------
// 4) h1 = relu(bn(t @ W1 + b1))     (T,128)x(128,128), row-guarded
// ---------------------------------------------------------------------------
__global__ __launch_bounds__(64) void tet_gemm1_kernel(
    const float* __restrict__ tin, const float* __restrict__ W,
    const float* __restrict__ b, const float* __restrict__ g,
    const float* __restrict__ be, float* __restrict__ out) {
  __shared__ __align__(16) float Alds[64 * LDA];
  __shared__ __align__(16) v2f WpBuf[2][8 * LDWP];
  const int tid = threadIdx.x, lane = tid & 31, w = tid >> 5;
  const int t0 = blockIdx.x * 64;

  for (int i = tid; i < 64 * (HDIM / 4); i += 64) {
    int r = i >> 5, c4 = (i & 31) << 2;
    int t = t0 + r;
    float4 v = make_float4(0.f, 0.f, 0.f, 0.f);
    if (t < T_TET) v = *(const float4*)(tin + (size_t)t * HDIM + c4);
    *(float4*)(Alds + r * LDA + c4) = v;
  }
  __syncthreads();

  v8f acc0[8], acc1[8];
  acc_zero2(acc0, acc1);
  gemm_main(Alds, W, HDIM, HDIM, LDA, WpBuf[0], WpBuf[1], tid, w, lane, acc0,
            acc1);

  const int mb = (lane >> 4) * 8;
  const int nc = lane & 15;
  const float inv_s = rsqrtf(1.0f + BN_EPS_F);
  float bias[8], gs[8], bev[8];
#pragma unroll
  for (int nt = 0; nt < 8; ++nt) {
    int cn = nt * 16 + nc;
    bias[nt] = b[cn];
    gs[nt]   = g[cn] * inv_s;
    bev[nt]  = be[cn];
  }
#pragma unroll
  for (int mt = 0; mt < 2; ++mt) {
    const v8f* acc = mt ? acc1 : acc0;
#pragma unroll
    for (int r = 0; r < 8; ++r) {
      int t = t0 + w * 32 + mt * 16 + mb + r;
      if (t < T_TET) {
        float* orow = out + (size_t)t * HDIM;
#pragma unroll
        for (int nt = 0; nt < 8; ++nt) {
          float v = (acc[nt][r] + bias[nt]) * gs[nt] + bev[nt];
          orow[nt * 16 + nc] = fmaxf(v, 0.0f);
        }
      }
    }
  }
}

// ---------------------------------------------------------------------------
// 5) tet = h1 @ W2 + b2; a[tetra_ids[t]] = tet[t]   (overwrite)
// ---------------------------------------------------------------------------
__global__ __launch_bounds__(64) void tet_gemm2_kernel(
    const float* __restrict__ tin, const float* __restrict__ W,
    const float* __restrict__ b, const int* __restrict__ tids,
    float* __restrict__ a_out) {
  __shared__ __align__(16) float Alds[64 * LDA];
  __shared__ __align__(16) v2f WpBuf[2][8 * LDWP];
  const int tid = threadIdx.x, lane = tid & 31, w = tid >> 5;
  const int t0 = blockIdx.x * 64;

  for (int i = tid; i < 64 * (HDIM / 4); i += 64) {
    int r = i >> 5, c4 = (i & 31) << 2;
    int t = t0 + r;
    float4 v = make_float4(0.f, 0.f, 0.f, 0.f);
    if (t < T_TET) v = *(const float4*)(tin + (size_t)t * HDIM + c4);
    *(float4*)(Alds + r * LDA + c4) = v;
  }
  __syncthreads();

  v8f acc0[8], acc1[8];
  acc_zero2(acc0, acc1);
  gemm_main(Alds, W, HDIM, HDIM, LDA, WpBuf[0], WpBuf[1], tid, w, lane, acc0,
            acc1);

  const int mb = (lane >> 4) * 8;
  const int nc = lane & 15;
  float bias[8];
#pragma unroll
  for (int nt = 0; nt < 8; ++nt) bias[nt] = b[nt * 16 + nc];
#pragma unroll
  for (int mt = 0; mt < 2; ++mt) {
    const v8f* acc = mt ? acc1 : acc0;
#pragma unroll
    for (int r = 0; r < 8; ++r) {
      int t = t0 + w * 32 + mt * 16 + mb + r;
      if (t < T_TET) {
        float* orow = a_out + (size_t)tids[t] * HDIM;
#pragma unroll
        for (int nt = 0; nt < 8; ++nt)
          orow[nt * 16 + nc] = acc[nt][r] + bias[nt];
      }
    }
  }
}

// ---------------------------------------------------------------------------
// 6) h_next = 2 * relu(bn((a[row[e]] - h[e^1]) @ mlp_W + mlp_b))
// ---------------------------------------------------------------------------
__global__ __launch_bounds__(64) void mlp_kernel(
    const float* __restrict__ h, const float* __restrict__ abuf,
    const int* __restrict__ rowA, const float* __restrict__ W,
    const float* __restrict__ b, const float* __restrict__ g,
    const float* __restrict__ be, float* __restrict__ out) {
  __shared__ __align__(16) float Alds[64 * LDA];
  __shared__ __align__(16) v2f WpBuf[2][8 * LDWP];
  const int tid = threadIdx.x, lane = tid & 31, w = tid >> 5;
  const int e0 = blockIdx.x * 64;

  for (int i = tid; i < 64 * (HDIM / 4); i += 64) {
    int r = i >> 5, c4 = (i & 31) << 2;
    int e = e0 + r;
    float4 av = *(const float4*)(abuf + (size_t)rowA[e] * HDIM + c4);
    float4 hv = *(const float4*)(h + (size_t)(e ^ 1) * HDIM + c4);
    float4 d;
    d.x = av.x - hv.x;
    d.y = av.y - hv.y;
    d.z = av.z - hv.z;
    d.w = av.w - hv.w;
    *(float4*)(Alds + r * LDA + c4) = d;
  }
  __syncthreads();

  v8f acc0[8], acc1[8];
  acc_zero2(acc0, acc1);
  gemm_main(Alds, W, HDIM, HDIM, LDA, WpBuf[0], WpBuf[1], tid, w, lane, acc0,
            acc1);

  const int mb = (lane >> 4) * 8;
  const int nc = lane & 15;
  const float inv_s = rsqrtf(1.0f + BN_EPS_F);
  float bias[8], gs[8], bev[8];
#pragma unroll
  for (int nt = 0; nt < 8; ++nt) {
    int cn = nt * 16 + nc;
    bias[nt] = b[cn];
    gs[nt]   = g[cn] * inv_s;
    bev[nt]  = be[cn];
  }
#pragma unroll
  for (int mt = 0; mt < 2; ++mt) {
    const v8f* acc = mt ? acc1 : acc0;
#pragma unroll
    for (int r = 0; r < 8; ++r) {
      int e = e0 + w * 32 + mt * 16 + mb + r;
      float* orow = out + (size_t)e * HDIM;
#pragma unroll
      for (int nt = 0; nt < 8; ++nt) {
        float v = (acc[nt][r] + bias[nt]) * gs[nt] + bev[nt];
        orow[nt * 16 + nc] = 2.0f * fmaxf(v, 0.0f);  // h = hh + e_h = 2*e_h
      }
    }
  }
}

// ---------------------------------------------------------------------------
// 7) pooling + FFN
// ---------------------------------------------------------------------------
__global__ void pool_accum_kernel(const float* __restrict__ a,
                                  const int* __restrict__ batch,
                                  float* __restrict__ pooled) {
  int idx = blockIdx.x * blockDim.x + threadIdx.x;
  if (idx >= N_NODES * HDIM) return;
  int n = idx >> 7, c = idx & 127;
  atomicAdd(pooled + (size_t)batch[n] * HDIM + c, a[idx]);
}

__global__ void pool_cnt_kernel(const int* __restrict__ batch,
                                float* __restrict__ cnt) {
  int n = blockIdx.x * blockDim.x + threadIdx.x;
  if (n < N_NODES) atomicAdd(cnt + batch[n], 1.0f);
}

__global__ void final_ffn_kernel(const float* __restrict__ pooled,
                                 const float* __restrict__ cnt,
                                 const float* __restrict__ fw,
                                 const float* __restrict__ fb,
                                 float* __restrict__ out) {
  __shared__ float red[128];
  int g = blockIdx.x, c = threadIdx.x;
  red[c] = (pooled[(size_t)g * HDIM + c] / cnt[g]) * fw[c];
  __syncthreads();
  for (int s = 64; s > 0; s >>= 1) {
    if (c < s) red[c] += red[c + s];
    __syncthreads();
  }
  if (c == 0) out[g] = red[0] + fb[0];
}

// ---------------------------------------------------------------------------
// Host orchestration
// ---------------------------------------------------------------------------
extern "C" void kernel_launch(void* const* d_in, const int* in_sizes, int n_in,
                              void* d_out, int out_size, void* d_ws,
                              size_t ws_size, hipStream_t stream) {
  (void)in_sizes; (void)n_in; (void)out_size; (void)ws_size;

  const float* x      = (const float*)d_in[0];
  const float* eattr  = (const float*)d_in[1];
  const int*   eindex = (const int*)d_in[2];
  const int*   batch  = (const int*)d_in[3];
  // d_in[4] parity_atoms: unused by the reference forward pass
  const int*   tids   = (const int*)d_in[5];
  const int*   te     = (const int*)d_in[6];
  const float* ei_W   = (const float*)d_in[7];
  const float* ei_b   = (const float*)d_in[8];
  const float* lin_W  = (const float*)d_in[9];
  const float* lin_b  = (const float*)d_in[10];
  const float* mlp_W  = (const float*)d_in[11];
  const float* mlp_b  = (const float*)d_in[12];
  const float* mlp_g  = (const float*)d_in[13];
  const float* mlp_be = (const float*)d_in[14];
  const float* tet_W1 = (const float*)d_in[15];
  const float* tet_b1 = (const float*)d_in[16];
  const float* tet_g  = (const float*)d_in[17];
  const float* tet_be = (const float*)d_in[18];
  const float* tet_W2 = (const float*)d_in[19];
  const float* tet_b2 = (const float*)d_in[20];
  const float* ffn_W  = (const float*)d_in[21];
  const float* ffn_b  = (const float*)d_in[22];

  const int* rowA = eindex;            // edge_index[0]
  const int* colA = eindex + E_EDGES;  // edge_index[1]

  char* ws = (char*)d_ws;
  size_t off = 0;
  auto carve = [&](size_t bytes) -> char* {
    char* p = ws + off;
    off += (bytes + 255) & ~(size_t)255;
    return p;
  };
  float* hA     = (float*)carve((size_t)E_EDGES * HDIM * sizeof(float));
  float* hB     = (float*)carve((size_t)E_EDGES * HDIM * sizeof(float));
  float* a_buf  = (float*)carve((size_t)N_NODES * HDIM * sizeof(float));
  float* t_buf  = (float*)carve((size_t)T_TET * HDIM * sizeof(float));
  float* t2_buf = (float*)carve((size_t)T_TET * HDIM * sizeof(float));
  float* pooled = (float*)carve((size_t)G_GRAPH * HDIM * sizeof(float));
  float* cnt    = (float*)carve((size_t)G_GRAPH * sizeof(float));

  const dim3 blk(64);
  const int eg_blocks = E_EDGES / 64;       // 3125
  const int tg_blocks = (T_TET + 63) / 64;  // 79
  const int HH = HDIM * HDIM;

  edge_init_kernel<<<eg_blocks, blk, 0, stream>>>(x, eattr, rowA, ei_W, ei_b, hA);

  float* hc = hA;
  float* hn = hB;
  for (int l = 0; l < NDEPTH; ++l) {
    hipMemsetAsync(a_buf, 0, (size_t)N_NODES * HDIM * sizeof(float), stream);
    msg_scatter_kernel<<<eg_blocks, blk, 0, stream>>>(
        hc, lin_W + (size_t)l * HH, lin_b + l * HDIM, colA, a_buf);
    tetra_t_kernel<<<(T_TET * HDIM + 255) / 256, 256, 0, stream>>>(hc, te, t_buf);
    tet_gemm1_kernel<<<tg_blocks, blk, 0, stream>>>(
        t_buf, tet_W1 + (size_t)l * HH, tet_b1 + l * HDIM, tet_g + l * HDIM,
        tet_be + l * HDIM, t2_buf);
    tet_gemm2_kernel<<<tg_blocks, blk, 0, stream>>>(
        t2_buf, tet_W2 + (size_t)l * HH, tet_b2 + l * HDIM, tids, a_buf);
    mlp_kernel<<<eg_blocks, blk, 0, stream>>>(
        hc, a_buf, rowA, mlp_W + (size_t)l * HH, mlp_b + l * HDIM,
        mlp_g + l * HDIM, mlp_be + l * HDIM, hn);
    float* tmp = hc; hc = hn; hn = tmp;
  }

  // Final conv (l = DEPTH): only a_msg is needed
  hipMemsetAsync(a_buf, 0, (size_t)N_NODES * HDIM * sizeof(float), stream);
  msg_scatter_kernel<<<eg_blocks, blk, 0, stream>>>(
      hc, lin_W + (size_t)NDEPTH * HH, lin_b + NDEPTH * HDIM, colA, a_buf);
  tetra_t_kernel<<<(T_TET * HDIM + 255) / 256, 256, 0, stream>>>(hc, te, t_buf);
  tet_gemm1_kernel<<<tg_blocks, blk, 0, stream>>>(
      t_buf, tet_W1 + (size_t)NDEPTH * HH, tet_b1 + NDEPTH * HDIM,
      tet_g + NDEPTH * HDIM, tet_be + NDEPTH * HDIM, t2_buf);
  tet_gemm2_kernel<<<tg_blocks, blk, 0, stream>>>(
      t2_buf, tet_W2 + (size_t)NDEPTH * HH, tet_b2 + NDEPTH * HDIM, tids, a_buf);

  hipMemsetAsync(pooled, 0, (size_t)G_GRAPH * HDIM * sizeof(float), stream);
  hipMemsetAsync(cnt, 0, (size_t)G_GRAPH * sizeof(float), stream);
  pool_accum_kernel<<<(N_NODES * HDIM + 255) / 256, 256, 0, stream>>>(a_buf, batch, pooled);
  pool_cnt_kernel<<<(N_NODES + 255) / 256, 256, 0, stream>>>(batch, cnt);
  final_ffn_kernel<<<G_GRAPH, 128, 0, stream>>>(pooled, cnt, ffn_W, ffn_b,
                                                (float*)d_out);
}